// NativeDeepseekYarnRope_22351009808480
// MI455X (gfx1250) — compile-verified
//
#include <hip/hip_runtime.h>
#include <stdint.h>

// ---------------------------------------------------------------------------
// YaRN (neox) RoPE for packed sequences on gfx1250.
//   q:  [T, 128, 64] bf16, k: [T, 1, 64] bf16, cache: [163840, 64] bf16
//   out = concat(q_out flat, k_out flat), bf16.
// Memory-bound: ~1.04 GB total traffic -> ~45us floor at 23.3 TB/s.
// One 512-thread block (16 wave32) per token; per thread: 2x b128 NT loads,
// native packed-bf16 rotate (v_pk_mul_bf16 / v_pk_add_bf16), 2x b128 NT
// stores. cos/sin row staged once per token into LDS via the Tensor Data
// Mover (tensor_load_to_lds + s_wait_tensorcnt), read back via ds_load_b128.
// ---------------------------------------------------------------------------

#define ROT  64
#define HALF 32
#define NH   128
#define BLOCK 512

typedef unsigned int   u32;
typedef unsigned short u16;
typedef u32 u32x4 __attribute__((ext_vector_type(4)));

// Native packed bf16 ops (VOP3P, CDNA5 §15.10 opcodes 42 / 35).
__device__ __forceinline__ u32 pk_mul_bf16(u32 a, u32 b) {
    u32 d;
    asm volatile("v_pk_mul_bf16 %0, %1, %2" : "=v"(d) : "v"(a), "v"(b));
    return d;
}
__device__ __forceinline__ u32 pk_add_bf16(u32 a, u32 b) {
    u32 d;
    asm volatile("v_pk_add_bf16 %0, %1, %2" : "=v"(d) : "v"(a), "v"(b));
    return d;
}

// One chunk of 8 rotary pairs, all math in packed bf16 (matches reference
// per-op bf16 rounding): out1 = x1*c + x2*(-s) ; out2 = x2*c + x1*s
__device__ __forceinline__ void rope_chunk(const u16* __restrict__ src,
                                           u16* __restrict__ dst,
                                           const u16* __restrict__ s_row,
                                           int c) {
    const u32x4 ca = *(const u32x4*)(s_row + c * 8);          // 8 cos bf16
    const u32x4 sa = *(const u32x4*)(s_row + HALF + c * 8);   // 8 sin bf16
    u32x4 a = __builtin_nontemporal_load((const u32x4*)(src + c * 8));        // x1
    u32x4 b = __builtin_nontemporal_load((const u32x4*)(src + HALF + c * 8)); // x2
    u32x4 o1, o2;
#pragma unroll
    for (int j = 0; j < 4; ++j) {
        const u32 ns = sa[j] ^ 0x80008000u;                   // -sin (both lanes)
        o1[j] = pk_add_bf16(pk_mul_bf16(a[j], ca[j]), pk_mul_bf16(b[j], ns));
        o2[j] = pk_add_bf16(pk_mul_bf16(b[j], ca[j]), pk_mul_bf16(a[j], sa[j]));
    }
    __builtin_nontemporal_store(o1, (u32x4*)(dst + c * 8));
    __builtin_nontemporal_store(o2, (u32x4*)(dst + HALF + c * 8));
}

__global__ __launch_bounds__(BLOCK)
void yarn_rope_kernel(const u16* __restrict__ q,
                      const u16* __restrict__ k,
                      const u16* __restrict__ cache,
                      const int* __restrict__ indptr,
                      const int* __restrict__ offsets,
                      u16* __restrict__ qo,
                      u16* __restrict__ ko,
                      int nseg) {
    __shared__ __align__(16) u16 s_row[ROT];   // staged bf16 cos|sin row

    const int tok = blockIdx.x;
    const int tid = threadIdx.x;

    // searchsorted(indptr, tok, 'right') - 1  (uniform -> simple SALU loop)
    int seg = 0;
#pragma unroll 1
    for (int i = 1; i < nseg; ++i) seg += (tok >= indptr[i]) ? 1 : 0;
    const int pos = offsets[seg] + (tok - indptr[seg]);
    const u16* crow = cache + (size_t)pos * ROT;

#if defined(__gfx1250__) && __has_builtin(__builtin_amdgcn_tensor_load_to_lds) && \
    __has_builtin(__builtin_amdgcn_s_wait_tensorcnt)
    // Stage the 128B cos/sin row via the Tensor Data Mover (wave 0 issues once;
    // TDM ignores EXEC). Descriptor per CDNA5 ISA. 6-arg form.
    if (tid < 32) {
        typedef u32 tdm4u __attribute__((ext_vector_type(4)));
        typedef int  tdm8i __attribute__((ext_vector_type(8)));
        typedef int  tdm4i __attribute__((ext_vector_type(4)));
        const uint64_t gaddr = (uint64_t)(uintptr_t)crow;
        const u32 laddr = (u32)(uintptr_t)(&s_row[0]);   // LDS byte addr = addr[31:0]
        tdm4u g0;
        g0[0] = 1u;                                      // count=1, user mode
        g0[1] = laddr;                                   // lds_addr
        g0[2] = (u32)gaddr;                              // global_addr[31:0]
        g0[3] = ((u32)(gaddr >> 32) & 0x01FFFFFFu)       // global_addr[56:32]
                | 0x80000000u;                           // type=2 ("image")
        tdm8i g1;
        g1[0] = (int)(1u << 16);          // workgroup_mask=0, data_size=1 (2 bytes)
        g1[1] = (int)(((u32)ROT) << 16);  // tensor_dim0 = 64 (low 16 @ bits[31:16])
        g1[2] = (int)(1u << 16);          // tensor_dim0 hi=0, tensor_dim1 = 1
        g1[3] = (int)(((u32)ROT) << 16);  // tensor_dim1 hi=0, tile_dim0 = 64
        g1[4] = 0;                        // tile_dim1 = 0 (unused), tile_dim2 = 0
        g1[5] = ROT;                      // tensor_dim0_stride = 64 elements
        g1[6] = 0;                        // stride hi / dim1_stride lo
        g1[7] = 0;                        // dim1_stride hi
        tdm4i z4 = {0, 0, 0, 0};          // groups 2/3 unused (<=2D tensor)
        tdm8i z8 = {0, 0, 0, 0, 0, 0, 0, 0};
        __builtin_amdgcn_tensor_load_to_lds(g0, g1, z4, z4, z8, 0);
        __builtin_amdgcn_s_wait_tensorcnt(0);            // TENSORcnt==0 before barrier
    }
#else
    if (tid < ROT) s_row[tid] = crow[tid];
#endif
    __syncthreads();

    // q: 128 heads x 4 chunks = 512 threads, exactly one chunk per thread.
    const int c = tid & 3;
    const int h = tid >> 2;
    {
        const size_t rowbase = ((size_t)tok * NH + h) * ROT;
        rope_chunk(q + rowbase, qo + rowbase, s_row, c);
    }
    // k: single head, 4 chunks handled by threads 0..3.
    if (tid < 4) {
        const size_t rowbase = (size_t)tok * ROT;
        rope_chunk(k + rowbase, ko + rowbase, s_row, tid);
    }
}

extern "C" void kernel_launch(void* const* d_in, const int* in_sizes, int n_in,
                              void* d_out, int out_size, void* d_ws, size_t ws_size,
                              hipStream_t stream) {
    const u16* q      = (const u16*)d_in[0];
    const u16* k      = (const u16*)d_in[1];
    const u16* cache  = (const u16*)d_in[2];
    const int* indptr = (const int*)d_in[3];
    const int* offs   = (const int*)d_in[4];

    const int T    = in_sizes[0] / (NH * ROT);   // tokens
    const int nseg = in_sizes[3] - 1;            // B segments (indptr has B+1)

    u16* qo = (u16*)d_out;
    u16* ko = qo + (size_t)T * NH * ROT;

    yarn_rope_kernel<<<dim3((unsigned)T), dim3(BLOCK), 0, stream>>>(
        q, k, cache, indptr, offs, qo, ko, nseg);
}